// SparseArch_55173149884529
// MI455X (gfx1250) — compile-verified
//
#include <hip/hip_runtime.h>

typedef __attribute__((ext_vector_type(2))) float v2f;
typedef __attribute__((ext_vector_type(8))) float v8f;

#define T_TOTAL   1048576           // jagged values per feature
#define NUM_U4    (T_TOTAL / 2)     // uint4 (=2 int64) chunks per feature
#define EMB_D     64
#define NB        1024              // streaming blocks
#define NT        256               // threads per streaming block (8 waves)

// ---------------------------------------------------------------------------
// Kernel 1: rowsum_k[j] = sum_d emb_k[j, d] via V_WMMA_F32_16X16X4_F32.
// rowsum = E (16x64) x ones(64x1); done as 16 chained 16x16x4 WMMAs with
// B = all-ones (B layout irrelevant when every element is 1.0).
// A layout (32-bit 16x4): lanes 0-15 -> K=k0,k0+1 ; lanes 16-31 -> K=k0+2,k0+3.
// C/D layout: VGPR r, lanes 0-15 -> M=r ; lanes 16-31 -> M=r+8.
// Launch: 1 block, 32 threads (one wave, EXEC all ones during WMMAs).
// ---------------------------------------------------------------------------
__global__ __launch_bounds__(32) void rowsum_wmma(const float* __restrict__ emb0,
                                                  const float* __restrict__ emb1,
                                                  float* __restrict__ rs) {
  const int lane  = threadIdx.x & 31;
  const int m     = lane & 15;
  const int khalf = (lane >> 4) << 1;   // 0 for lanes 0-15, 2 for lanes 16-31

  v2f bones; bones[0] = 1.0f; bones[1] = 1.0f;
  v8f c0 = {};   // emb_0 rows 0..15
  v8f c1 = {};   // emb_1 rows 0..15
  v8f c2 = {};   // emb_1 rows 16..31

  for (int k0 = 0; k0 < EMB_D; k0 += 4) {
    const int kk = k0 + khalf;
    v2f a0, a1, a2;
    a0[0] = emb0[m * EMB_D + kk];            a0[1] = emb0[m * EMB_D + kk + 1];
    a1[0] = emb1[m * EMB_D + kk];            a1[1] = emb1[m * EMB_D + kk + 1];
    a2[0] = emb1[(m + 16) * EMB_D + kk];     a2[1] = emb1[(m + 16) * EMB_D + kk + 1];
    c0 = __builtin_amdgcn_wmma_f32_16x16x4_f32(false, a0, false, bones, (short)0, c0, false, false);
    c1 = __builtin_amdgcn_wmma_f32_16x16x4_f32(false, a1, false, bones, (short)0, c1, false, false);
    c2 = __builtin_amdgcn_wmma_f32_16x16x4_f32(false, a2, false, bones, (short)0, c2, false, false);
  }

  // D[m][n] == rowsum[m] for every n; read column n=0 from lane 0 / lane 16.
  if (lane == 0) {
    for (int r = 0; r < 8; ++r) {
      rs[r]      = c0[r];      // rowsum0[0..7]
      rs[16 + r] = c1[r];      // rowsum1[0..7]
      rs[32 + r] = c2[r];      // rowsum1[16..23]
    }
  } else if (lane == 16) {
    for (int r = 0; r < 8; ++r) {
      rs[8 + r]  = c0[r];      // rowsum0[8..15]
      rs[24 + r] = c1[r];      // rowsum1[8..15]
      rs[40 + r] = c2[r];      // rowsum1[24..31]
    }
  }
}

// ---------------------------------------------------------------------------
// Kernel 2: bandwidth-bound stream over both value arrays (int64 loaded as
// uint4 = 2 values). acc += rowsum[value & (zch-1)] via tiny LDS tables.
// Fixed-order LDS tree reduction -> one partial per block (deterministic).
// ---------------------------------------------------------------------------
__global__ __launch_bounds__(NT) void pool_stream(const uint4* __restrict__ v0,
                                                  const uint4* __restrict__ v1,
                                                  const float* __restrict__ rs,
                                                  float* __restrict__ partials) {
  __shared__ float srs0[16];
  __shared__ float srs1[32];
  __shared__ float red[NT];

  const int t = threadIdx.x;
  if (t < 16)      srs0[t]      = rs[t];
  else if (t < 48) srs1[t - 16] = rs[t];
  __syncthreads();

  float acc = 0.0f;
  const int stride = NB * NT;
  for (int i = blockIdx.x * NT + t; i < NUM_U4; i += stride) {
    const int inext = i + stride;
    if (inext < NUM_U4) {                 // emits global_prefetch_b8
      __builtin_prefetch(v0 + inext, 0, 0);
      __builtin_prefetch(v1 + inext, 0, 0);
    }
    uint4 a = v0[i];                      // two int64 values (low dwords x, z)
    uint4 b = v1[i];
    acc += srs0[a.x & 15u];
    acc += srs0[a.z & 15u];
    acc += srs1[b.x & 31u];
    acc += srs1[b.z & 31u];
  }

  red[t] = acc;
  __syncthreads();
  for (int s = NT >> 1; s > 0; s >>= 1) {
    if (t < s) red[t] += red[t + s];
    __syncthreads();
  }
  if (t == 0) partials[blockIdx.x] = red[0];
}

// ---------------------------------------------------------------------------
// Kernel 3: reduce NB partials, scale by 1/(B*2*D) = 2^-24 (exact in fp32).
// ---------------------------------------------------------------------------
__global__ __launch_bounds__(NB) void finalize_mean(const float* __restrict__ partials,
                                                    float* __restrict__ out) {
  __shared__ float red[NB];
  const int t = threadIdx.x;
  red[t] = partials[t];
  __syncthreads();
  for (int s = NB >> 1; s > 0; s >>= 1) {
    if (t < s) red[t] += red[t + s];
    __syncthreads();
  }
  if (t == 0) out[0] = red[0] * (1.0f / 16777216.0f);
}

// ---------------------------------------------------------------------------
// Inputs (setup_inputs order):
//   0: values_0      int64 [T]      -> streamed as uint4
//   1: segment_ids_0 int64 [T]      -> UNUSED (mean collapses segments)
//   2: values_1      int64 [T]      -> streamed as uint4
//   3: segment_ids_1 int64 [T]      -> UNUSED
//   4: emb_0         f32 [16*64]
//   5: emb_1         f32 [32*64]
// Output: 1 float (loss).
// Workspace: floats: [0..47] rowsums, [64..64+NB) block partials.
// ---------------------------------------------------------------------------
extern "C" void kernel_launch(void* const* d_in, const int* in_sizes, int n_in,
                              void* d_out, int out_size, void* d_ws, size_t ws_size,
                              hipStream_t stream) {
  (void)in_sizes; (void)n_in; (void)out_size; (void)ws_size;

  const uint4* v0   = (const uint4*)d_in[0];
  const uint4* v1   = (const uint4*)d_in[2];
  const float* emb0 = (const float*)d_in[4];
  const float* emb1 = (const float*)d_in[5];

  float* ws       = (float*)d_ws;
  float* rs       = ws;        // 48 floats
  float* partials = ws + 64;   // NB floats

  rowsum_wmma<<<1, 32, 0, stream>>>(emb0, emb1, rs);
  pool_stream<<<NB, NT, 0, stream>>>(v0, v1, rs, partials);
  finalize_mean<<<1, NB, 0, stream>>>(partials, (float*)d_out);
}